// GATNet_59030030516799
// MI455X (gfx1250) — compile-verified
//
#include <hip/hip_runtime.h>
#include <hip/hip_bf16.h>

// ---------------- problem constants (from reference) ----------------
#define N_NODES 100000
#define N_EDGES 1600000
#define ETOT    (N_EDGES + N_NODES)
#define M_SEG   20000
#define G_SEG   512
#define F_IN    64
#define F_ED    16
#define F_OUT   64
#define N_H     4
#define TPB     256

typedef __attribute__((ext_vector_type(2))) float v2f;
typedef __attribute__((ext_vector_type(8))) float v8f;

__device__ __forceinline__ float lrelu(float v, float s) { return v > 0.f ? v : s * v; }

// ---------------- WMMA f32 GEMM: C[M,N] = A[M,K] @ B[K,N], row-major ----------------
// One 16x16 output tile per wave32, V_WMMA_F32_16X16X4_F32 along K.
// Requires M%16==0, N%16==0, K%4==0 (all true for this net).
__global__ void gemm_wmma_f32(const float* __restrict__ A, const float* __restrict__ B,
                              float* __restrict__ C, int Mdim, int Kdim, int Ndim) {
    int wave = blockIdx.x * (blockDim.x >> 5) + (threadIdx.x >> 5);
    int lane = threadIdx.x & 31;
    int ntN  = Ndim >> 4;
    int ntM  = Mdim >> 4;
    if (wave >= ntM * ntN) return;
    int mt = wave / ntN;
    int nt = wave % ntN;

    int arow  = mt * 16 + (lane & 15);          // A: lanes 0-15 / 16-31 both cover M=0..15
    int bcol  = nt * 16 + (lane & 15);          // B/C: row striped across lanes
    int khalf = (lane >> 4) * 2;                // lanes 16-31 hold K+2,K+3

    const float* __restrict__ Ar = A + arow * Kdim + khalf;
    const float* __restrict__ Bc = B + khalf * Ndim + bcol;

    v8f acc = {};
    for (int k = 0; k < Kdim; k += 4) {
        v2f a, b;
        a.x = Ar[k];
        a.y = Ar[k + 1];
        b.x = Bc[k * Ndim];
        b.y = Bc[(k + 1) * Ndim];
        acc = __builtin_amdgcn_wmma_f32_16x16x4_f32(false, a, false, b, (short)0, acc,
                                                    false, false);
    }
    // C/D layout: VGPR i -> row mt*16 + i (+8 for lanes 16-31), col = bcol
    int orow = mt * 16 + ((lane >> 4) << 3);
#pragma unroll
    for (int i = 0; i < 8; ++i)
        C[(orow + i) * Ndim + bcol] = acc[i];
}

// ---------------- utility ----------------
// vectorized zero: n must be a multiple of 4 (all our buffers are)
__global__ void k_zero4(float4* __restrict__ p, int n4) {
    int i = blockIdx.x * blockDim.x + threadIdx.x;
    if (i < n4) p[i] = make_float4(0.f, 0.f, 0.f, 0.f);
}

// mean of edge_attr rows -> ea_mean[16]  (ea_mean must be pre-zeroed)
__global__ void k_ea_mean(const float* __restrict__ ea, float* __restrict__ ea_mean) {
    __shared__ float sm[F_ED];
    float loc[F_ED];
#pragma unroll
    for (int d = 0; d < F_ED; ++d) loc[d] = 0.f;
    for (int e = blockIdx.x * blockDim.x + threadIdx.x; e < N_EDGES;
         e += gridDim.x * blockDim.x) {
        const float* r = ea + e * F_ED;
#pragma unroll
        for (int d = 0; d < F_ED; ++d) loc[d] += r[d];
    }
    if (threadIdx.x < F_ED) sm[threadIdx.x] = 0.f;
    __syncthreads();
#pragma unroll
    for (int d = 0; d < F_ED; ++d) atomicAdd(&sm[d], loc[d]);
    __syncthreads();
    if (threadIdx.x < F_ED)
        atomicAdd(&ea_mean[threadIdx.x], sm[threadIdx.x] * (1.f / (float)N_EDGES));
}

// Per-layer: Ve[d*4+h] = sum_c We[d,h*C+c]*ae[h*C+c];  selfAE[h] = ea_mean . Ve[:,h]
__global__ void k_prep(const float* __restrict__ We, const float* __restrict__ ae,
                       const float* __restrict__ ea_mean, float* __restrict__ Ve,
                       float* __restrict__ selfAE, int HC, int C) {
    __shared__ float sVe[F_ED * N_H];
    int t = threadIdx.x;                 // 0..63
    int d = t >> 2, h = t & 3;
    float s = 0.f;
    for (int c = 0; c < C; ++c) s += We[d * HC + h * C + c] * ae[h * C + c];
    sVe[d * N_H + h] = s;
    Ve[d * N_H + h]  = s;
    __syncthreads();
    if (t < N_H) {
        float q = 0.f;
        for (int dd = 0; dd < F_ED; ++dd) q += ea_mean[dd] * sVe[dd * N_H + t];
        selfAE[t] = q;
    }
}

// alphaE[e*4+h] = ea[e,:] . Ve[:,h]   (real edges only)
__global__ void k_alpha_e(const float* __restrict__ ea, const float* __restrict__ Ve,
                          float* __restrict__ alphaE) {
    int i = blockIdx.x * blockDim.x + threadIdx.x;
    if (i >= N_EDGES * N_H) return;
    int e = i >> 2, h = i & 3;
    const float* r = ea + e * F_ED;
    float s = 0.f;
#pragma unroll
    for (int d = 0; d < F_ED; ++d) s += r[d] * Ve[d * N_H + h];
    alphaE[i] = s;
}

// per-node attention logits: alpha_s[n,h] = hx[n,h,:].a_s[h,:]; same for a_d
__global__ void k_node_alpha(const float* __restrict__ hx, const float* __restrict__ a_s,
                             const float* __restrict__ a_d, float* __restrict__ alpha_s,
                             float* __restrict__ alpha_d, int HC, int C) {
    int i = blockIdx.x * blockDim.x + threadIdx.x;
    if (i >= N_NODES * N_H) return;
    int n = i >> 2, h = i & 3;
    const float* hr = hx + n * HC + h * C;
    float ss = 0.f, sd = 0.f;
    for (int c = 0; c < C; ++c) {
        ss += hr[c] * a_s[h * C + c];
        sd += hr[c] * a_d[h * C + c];
    }
    alpha_s[i] = ss;
    alpha_d[i] = sd;
}

__device__ __forceinline__ unsigned fenc(float f) {
    unsigned b = __float_as_uint(f);
    return (b & 0x80000000u) ? ~b : (b | 0x80000000u);
}
__device__ __forceinline__ float fdec(unsigned u) {
    unsigned b = (u & 0x80000000u) ? (u & 0x7fffffffu) : ~u;
    return __uint_as_float(b);
}

// pass1: raw logit (leaky 0.2), atomic segment-max over dst (monotone-uint trick)
__global__ void k_pass1(const int* __restrict__ srcA, const int* __restrict__ dstA,
                        const float* __restrict__ alpha_s, const float* __restrict__ alpha_d,
                        const float* __restrict__ alphaE, const float* __restrict__ selfAE,
                        float* __restrict__ raw, unsigned* __restrict__ amax) {
    int i = blockIdx.x * blockDim.x + threadIdx.x;
    if (i >= ETOT * N_H) return;
    int e = i >> 2, h = i & 3;
    int s, d;
    float aE;
    if (e < N_EDGES) { s = srcA[e]; d = dstA[e]; aE = alphaE[i]; }
    else             { s = d = e - N_EDGES;      aE = selfAE[h]; }
    float v = alpha_s[s * N_H + h] + alpha_d[d * N_H + h] + aE;
    v = lrelu(v, 0.2f);
    raw[i] = v;
    atomicMax(&amax[d * N_H + h], fenc(v));
}

// pass2: ex = exp(raw - max[dst]); atomic segment-sum
__global__ void k_pass2(const int* __restrict__ dstA, const unsigned* __restrict__ amax,
                        float* __restrict__ raw_ex, float* __restrict__ den) {
    int i = blockIdx.x * blockDim.x + threadIdx.x;
    if (i >= ETOT * N_H) return;
    int e = i >> 2, h = i & 3;
    int d = (e < N_EDGES) ? dstA[e] : (e - N_EDGES);
    float mx = fdec(amax[d * N_H + h]);
    float ev = __expf(raw_ex[i] - mx);
    raw_ex[i] = ev;
    atomicAdd(&den[d * N_H + h], ev);
}

// pass3: out[dst] += hx[src,h,:] * ex/(den[dst]+eps) * scale  (1/den folded here)
__global__ void k_pass3(const int* __restrict__ srcA, const int* __restrict__ dstA,
                        const float* __restrict__ ex, const float* __restrict__ den,
                        const float* __restrict__ hx, float* __restrict__ outB,
                        int HC, int C, int outW, int concat, float scale) {
    int i = blockIdx.x * blockDim.x + threadIdx.x;
    if (i >= ETOT * N_H) return;
    int e = i >> 2, h = i & 3;
    int s, d;
    if (e < N_EDGES) { s = srcA[e]; d = dstA[e]; }
    else             { s = d = e - N_EDGES; }
    const float* hr = hx + s * HC + h * C;
    // start the random-gather row moving (global_prefetch_b8) before the
    // dependent ex/den loads; the row is L2-resident, this pulls it to WGP$
    __builtin_prefetch(hr, 0, 0);
    float w = ex[i] / (den[d * N_H + h] + 1e-16f) * scale;
    float* orow = outB + d * outW + (concat ? h * C : 0);
    for (int c = 0; c < C; ++c) atomicAdd(&orow[c], hr[c] * w);
}

// bias + outer leaky-relu 0.01 (in place); Wm1 = width-1 (width is a power of two)
__global__ void k_finalize(float* __restrict__ outB, const float* __restrict__ bias, int Wm1,
                           int total) {
    int i = blockIdx.x * blockDim.x + threadIdx.x;
    if (i >= total) return;
    outB[i] = lrelu(outB[i] + bias[i & Wm1], 0.01f);
}

// pooling: pool[scatter_index[n], :] += h4[n, :]
__global__ void k_pool_scatter(const float* __restrict__ h4, const int* __restrict__ sidx,
                               float* __restrict__ pool) {
    int i = blockIdx.x * blockDim.x + threadIdx.x;
    if (i >= N_NODES * F_OUT) return;
    int n = i >> 6, c = i & 63;
    atomicAdd(&pool[sidx[n] * F_OUT + c], h4[i]);
}

// poolG[nodes[m], :] += pool[m, :] / scatter_norm[m]
__global__ void k_pool_norm(const float* __restrict__ pool, const float* __restrict__ snorm,
                            const int* __restrict__ nodes, float* __restrict__ poolG) {
    int i = blockIdx.x * blockDim.x + threadIdx.x;
    if (i >= M_SEG * F_OUT) return;
    int m = i >> 6, c = i & 63;
    atomicAdd(&poolG[nodes[m] * F_OUT + c], pool[i] / snorm[m]);
}

__global__ void k_tgt_gather(const float* __restrict__ h4, const int* __restrict__ dvi,
                             float* __restrict__ tgt) {
    int i = blockIdx.x * blockDim.x + threadIdx.x;
    if (i >= G_SEG * F_OUT) return;
    int g = i >> 6;
    tgt[i] = h4[dvi[g] * F_OUT + (i & 63)];
}

// out[g] = bo + sum_j lrelu(featA[g,j])*Wo[j] + sum_j lrelu(featB[g,j])*Wo[64+j]
__global__ void k_out(const float* __restrict__ featA, const float* __restrict__ featB,
                      const float* __restrict__ Wo, const float* __restrict__ bo,
                      float* __restrict__ out) {
    int g = blockIdx.x * blockDim.x + threadIdx.x;
    if (g >= G_SEG) return;
    float acc = bo[0];
#pragma unroll 4
    for (int j = 0; j < F_OUT; ++j) acc += lrelu(featA[g * F_OUT + j], 0.01f) * Wo[j];
#pragma unroll 4
    for (int j = 0; j < F_OUT; ++j) acc += lrelu(featB[g * F_OUT + j], 0.01f) * Wo[F_OUT + j];
    out[g] = acc;
}

// ---------------- host orchestration ----------------
static inline int cdiv(long long a, int b) { return (int)((a + b - 1) / b); }

static inline void zero_buf(float* p, size_t n, hipStream_t stream) {
    int n4 = (int)(n >> 2);
    k_zero4<<<cdiv(n4, TPB), TPB, 0, stream>>>((float4*)p, n4);
}

extern "C" void kernel_launch(void* const* d_in, const int* in_sizes, int n_in,
                              void* d_out, int out_size, void* d_ws, size_t ws_size,
                              hipStream_t stream) {
    const float* x   = (const float*)d_in[0];
    const int*   ei  = (const int*)d_in[1];           // [2,E]
    const float* ea  = (const float*)d_in[2];
    const int*   sidx = (const int*)d_in[3];
    const float* snorm = (const float*)d_in[4];
    const int*   nodes = (const int*)d_in[5];
    const int*   dvi   = (const int*)d_in[6];
    const float* Wp = (const float*)d_in[31];
    const float* Wt = (const float*)d_in[32];
    const float* Wo = (const float*)d_in[33];
    const float* bo = (const float*)d_in[34];
    float* out = (float*)d_out;

    const int* srcA = ei;
    const int* dstA = ei + N_EDGES;

    // ---- workspace layout (floats) ----
    float* base = (float*)d_ws;
    size_t off = 0;
    auto alloc = [&](size_t n) { size_t o = off; off += (n + 63) & ~(size_t)63; return base + o; };
    float* cbuf   = alloc(128);                 // [0..15] ea_mean, [16..79] Ve, [80..83] selfAE
    float* hx     = alloc((size_t)N_NODES * 256);
    float* hbuf0  = alloc((size_t)N_NODES * 64);
    float* hbuf1  = alloc((size_t)N_NODES * 64);
    float* alphaS = alloc((size_t)N_NODES * N_H);
    float* alphaD = alloc((size_t)N_NODES * N_H);
    float* amaxF  = alloc((size_t)N_NODES * N_H);   // used as unsigned
    float* den    = alloc((size_t)N_NODES * N_H);
    float* alphaE = alloc((size_t)N_EDGES * N_H);
    float* exbuf  = alloc((size_t)ETOT * N_H);
    float* pool   = alloc((size_t)M_SEG * F_OUT);
    float* poolG  = alloc((size_t)G_SEG * F_OUT);
    float* tgt    = alloc((size_t)G_SEG * F_OUT);
    float* featA  = alloc((size_t)G_SEG * F_OUT);
    float* featB  = alloc((size_t)G_SEG * F_OUT);
    (void)ws_size; (void)n_in; (void)in_sizes; (void)out_size;

    float* ea_mean = cbuf;
    float* Ve      = cbuf + 16;
    float* selfAE  = cbuf + 80;

    // ---- edge-attr mean ----
    zero_buf(cbuf, 128, stream);
    k_ea_mean<<<512, TPB, 0, stream>>>(ea, ea_mean);

    // ---- per-layer configs ----
    // layer l: input (ptr,width), W,We,asrc,adst,ae,b at d_in[7+6l .. 12+6l]
    const float* hin[4]  = { x, hbuf0, hbuf1, hbuf0 };
    float*       hout[4] = { hbuf0, hbuf1, hbuf0, hbuf1 };
    const int Cin[4]  = { F_IN, 32, 32, 32 };
    const int Cc[4]   = { 8, 8, 8, 64 };
    const int HC[4]   = { 32, 32, 32, 256 };
    const int outW[4] = { 32, 32, 32, 64 };
    const int concat[4] = { 1, 1, 1, 0 };
    const float scale[4] = { 1.f, 1.f, 1.f, 1.f / (float)N_H };

    const int edgeT = ETOT * N_H;
    for (int l = 0; l < 4; ++l) {
        const float* W  = (const float*)d_in[7 + 6 * l];
        const float* We = (const float*)d_in[8 + 6 * l];
        const float* as = (const float*)d_in[9 + 6 * l];
        const float* ad = (const float*)d_in[10 + 6 * l];
        const float* ae_l = (const float*)d_in[11 + 6 * l];
        const float* b  = (const float*)d_in[12 + 6 * l];

        // hx = hin @ W   (WMMA)
        {
            int waves = (N_NODES / 16) * (HC[l] / 16);
            gemm_wmma_f32<<<cdiv(waves, 8), TPB, 0, stream>>>(hin[l], W, hx, N_NODES,
                                                              Cin[l], HC[l]);
        }
        k_prep<<<1, 64, 0, stream>>>(We, ae_l, ea_mean, Ve, selfAE, HC[l], Cc[l]);
        k_alpha_e<<<cdiv((long long)N_EDGES * N_H, TPB), TPB, 0, stream>>>(ea, Ve, alphaE);
        k_node_alpha<<<cdiv((long long)N_NODES * N_H, TPB), TPB, 0, stream>>>(
            hx, as, ad, alphaS, alphaD, HC[l], Cc[l]);

        zero_buf(amaxF, (size_t)N_NODES * N_H, stream);
        zero_buf(den, (size_t)N_NODES * N_H, stream);
        zero_buf(hout[l], (size_t)N_NODES * outW[l], stream);

        k_pass1<<<cdiv(edgeT, TPB), TPB, 0, stream>>>(srcA, dstA, alphaS, alphaD, alphaE,
                                                      selfAE, exbuf, (unsigned*)amaxF);
        k_pass2<<<cdiv(edgeT, TPB), TPB, 0, stream>>>(dstA, (const unsigned*)amaxF, exbuf, den);
        k_pass3<<<cdiv(edgeT, TPB), TPB, 0, stream>>>(srcA, dstA, exbuf, den, hx, hout[l],
                                                      HC[l], Cc[l], outW[l], concat[l],
                                                      scale[l]);
        k_finalize<<<cdiv((long long)N_NODES * outW[l], TPB), TPB, 0, stream>>>(
            hout[l], b, outW[l] - 1, N_NODES * outW[l]);
    }

    float* h4 = hbuf1;  // layer-4 output, [N,64]

    // ---- pooling ----
    zero_buf(pool, (size_t)M_SEG * F_OUT, stream);
    zero_buf(poolG, (size_t)G_SEG * F_OUT, stream);
    k_pool_scatter<<<cdiv((long long)N_NODES * F_OUT, TPB), TPB, 0, stream>>>(h4, sidx, pool);
    k_pool_norm<<<cdiv((long long)M_SEG * F_OUT, TPB), TPB, 0, stream>>>(pool, snorm, nodes,
                                                                         poolG);
    k_tgt_gather<<<cdiv((long long)G_SEG * F_OUT, TPB), TPB, 0, stream>>>(h4, dvi, tgt);

    // ---- head: two WMMA GEMMs + fused output matvec ----
    {
        int waves = (G_SEG / 16) * (F_OUT / 16);
        gemm_wmma_f32<<<cdiv(waves, 8), TPB, 0, stream>>>(poolG, Wp, featA, G_SEG, F_OUT, F_OUT);
        gemm_wmma_f32<<<cdiv(waves, 8), TPB, 0, stream>>>(tgt, Wt, featB, G_SEG, F_OUT, F_OUT);
    }
    k_out<<<cdiv(G_SEG, TPB), TPB, 0, stream>>>(featA, featB, Wo, bo, out);
}